// SphKernel_65309272703426
// MI455X (gfx1250) — compile-verified
//
#include <hip/hip_runtime.h>
#include <hip/hip_bf16.h>

typedef __attribute__((ext_vector_type(2))) float v2f;
typedef __attribute__((ext_vector_type(4))) float v4f;
typedef __attribute__((ext_vector_type(8))) float v8f;

// One lane per point. P=64 and blockDim=256 guarantee each wave32's 32 points
// share a single (b,v) frame, and N % 256 == 0 so EXEC is always all-ones
// (required for WMMA).
__global__ __launch_bounds__(256) void
SphKernel_65309272703426_kernel(const float* __restrict__ patches,
                                const float* __restrict__ frames,
                                const float* __restrict__ rr,
                                float* __restrict__ out)
{
    const int pid  = blockIdx.x * blockDim.x + threadIdx.x;  // point id
    const int lane = threadIdx.x & 31;
    const int m    = lane & 15;
    const bool hi  = lane >= 16;

    // ---- load patch vector and normalize (eps matches tf l2_normalize) ----
    const float px = patches[pid * 3 + 0];
    const float py = patches[pid * 3 + 1];
    const float pz = patches[pid * 3 + 2];
    const float sq  = px * px + py * py + pz * pz;
    const float inv = rsqrtf(fmaxf(sq, 1e-12f));
    const float dx = px * inv, dy = py * inv, dz = pz * inv;

    // ---- per-(b,v) frame: wave-uniform -> scalar loads via readfirstlane ----
    const int bv  = pid >> 6;                          // /P (P=64)
    const int bvs = __builtin_amdgcn_readfirstlane(bv);
    const float* F = frames + (size_t)bvs * 9;         // F[i*3+j] = frames[i][j]
    const float f00 = F[0], f01 = F[1], f02 = F[2];
    const float f10 = F[3], f11 = F[4], f12 = F[5];
    const float f20 = F[6], f21 = F[7], f22 = F[8];

    // ---- A operands: A[M=j][K=i] = frames[i][j] (so D = Ft*dirs = proj) ----
    // 16x4 f32 A layout: lanes 0-15 hold (K=0,K=1), lanes 16-31 hold (K=2,K=3).
    const float g0 = hi ? f20 : f00;   // K=0 or K=2 column, row j=0
    const float g1 = hi ? f21 : f01;   // j=1
    const float g2 = hi ? f22 : f02;   // j=2
    v2f A1, A2;
    // WMMA #1: Ft in rows 0..2  -> results for lanes 0-15 (D rows 0..2)
    A1.x = (m == 0) ? g0 : (m == 1) ? g1 : (m == 2) ? g2 : 0.0f;
    A1.y = hi ? 0.0f : ((m == 0) ? f10 : (m == 1) ? f11 : (m == 2) ? f12 : 0.0f);
    // WMMA #2: Ft in rows 8..10 -> results for lanes 16-31 (D rows 8..10)
    A2.x = (m == 8) ? g0 : (m == 9) ? g1 : (m == 10) ? g2 : 0.0f;
    A2.y = hi ? 0.0f : ((m == 8) ? f10 : (m == 9) ? f11 : (m == 10) ? f12 : 0.0f);

    // ---- B operands: B[K=i][N=p] = dirs[p][i] ----
    // 4x16 f32 B layout mirrors A: lanes 0-15 hold (K=0,K=1), 16-31 hold (K=2,K=3).
    const int src1 = m;            // points 0..15 of the wave
    const int src2 = 16 + m;       // points 16..31 of the wave
    const float x1 = __shfl(dx, src1), y1 = __shfl(dy, src1), z1 = __shfl(dz, src1);
    const float x2s = __shfl(dx, src2), y2s = __shfl(dy, src2), z2s = __shfl(dz, src2);
    v2f B1, B2;
    B1.x = hi ? z1 : x1;   B1.y = hi ? 0.0f : y1;
    B2.x = hi ? z2s : x2s; B2.y = hi ? 0.0f : y2s;

    // ---- rotate: two f32 WMMAs, each lane recovers its own point's proj ----
    v8f czero = {};
    v8f d1 = __builtin_amdgcn_wmma_f32_16x16x4_f32(false, A1, false, B1,
                                                   (short)0, czero, false, false);
    v8f d2 = __builtin_amdgcn_wmma_f32_16x16x4_f32(false, A2, false, B2,
                                                   (short)0, czero, false, false);
    // lanes<16: d1[k]=row k (proj), d2[k]=row k of zeros; lanes>=16: d1[k]=row k+8
    // (zeros), d2[k]=row k+8 (proj). Sum hands every lane its own (x,y,z).
    const float X = d1[0] + d2[0];
    const float Y = d1[1] + d2[1];
    const float Z = d1[2] + d2[2];

    // ---- real spherical harmonics up to l=3, reference channel order ----
    const float x2 = X * X, y2 = Y * Y, z2 = Z * Z;
    const float xy = X * Y, yz = Y * Z, zx = Z * X;
    float Yc[16];
    Yc[0]  = 0.28209479177387814f;                                   // Y00
    Yc[1]  = 0.4886025119029199f * Z;                                // Y100
    Yc[2]  = 0.31539156525252005f * (2.0f * z2 - x2 - y2);           // Y200
    Yc[3]  = 0.3731763325901154f * (2.0f * z2 * Z - 3.0f * x2 * Z - 3.0f * y2 * Z); // Y300
    Yc[4]  = 0.4886025119029199f * X;                                // Y1_10
    Yc[5]  = 0.4886025119029199f * Y;                                // Y1_11
    Yc[6]  = 1.0925484305920792f * zx;                               // Y2_10
    Yc[7]  = 1.0925484305920792f * yz;                               // Y2_11
    Yc[8]  = 0.4570457994644658f * (4.0f * z2 * X - x2 * X - X * y2);  // Y3_10
    Yc[9]  = 0.4570457994644658f * (4.0f * Y * z2 - x2 * Y - y2 * Y);  // Y3_11
    Yc[10] = 0.5462742152960396f * (x2 - y2);                        // Y2_20
    Yc[11] = 1.0925484305920792f * xy;                               // Y2_21
    Yc[12] = 1.445305721320277f * (x2 * Z - y2 * Z);                 // Y3_20
    Yc[13] = 2.890611442640554f * xy * Z;                            // Y3_21
    Yc[14] = 0.5900435899266435f * (x2 * X - 3.0f * X * y2);         // Y3_30
    Yc[15] = 0.5900435899266435f * (3.0f * x2 * Y - y2 * Y);         // Y3_31

    // ---- gaussian shells: centers {0, 0.5, 1}, k = 4*ln(0.5) ----
    const float rv = rr[pid];
    const float kg = -2.772588722239781f;
    const float e0 = rv, e1 = rv - 0.5f, e2 = rv - 1.0f;
    float gs0 = __expf(kg * e0 * e0);
    float gs1 = __expf(kg * e1 * e1);
    float gs2 = __expf(kg * e2 * e2);
    const float gn = 1.0f / (gs0 + gs1 + gs2);
    gs0 *= gn; gs1 *= gn; gs2 *= gn;
    const float gsv[3] = {gs0, gs1, gs2};

    // ---- outer product, 12x b128 non-temporal stores (201MB stream > L2) ----
    v4f* o = reinterpret_cast<v4f*>(out + (size_t)pid * 48);
#pragma unroll
    for (int s = 0; s < 3; ++s) {
        const float g = gsv[s];
#pragma unroll
        for (int q = 0; q < 4; ++q) {
            v4f v;
            v.x = g * Yc[4 * q + 0];
            v.y = g * Yc[4 * q + 1];
            v.z = g * Yc[4 * q + 2];
            v.w = g * Yc[4 * q + 3];
            __builtin_nontemporal_store(v, o + s * 4 + q);
        }
    }
}

extern "C" void kernel_launch(void* const* d_in, const int* in_sizes, int n_in,
                              void* d_out, int out_size, void* d_ws, size_t ws_size,
                              hipStream_t stream) {
    const float* patches = (const float*)d_in[0];  // [B,V,P,3]
    const float* frames  = (const float*)d_in[1];  // [B,V,3,3]
    const float* r       = (const float*)d_in[2];  // [B,V,P]
    float* out = (float*)d_out;                    // [B,V,P,3,16]

    const int N = in_sizes[2];            // B*V*P points (1,048,576; % 256 == 0)
    const int block = 256;                // 8 wave32s per workgroup
    const int grid  = N / block;
    SphKernel_65309272703426_kernel<<<grid, block, 0, stream>>>(patches, frames, r, out);
}